// ModelNew_43173011259797
// MI455X (gfx1250) — compile-verified
//
#include <hip/hip_runtime.h>

#define BB 4
#define SS 4096
#define HH 16
#define PP 64
#define NN 64
#define LL 64
#define CH 64          // number of chunks
#define PAD 68         // LDS row stride (floats): keeps 16B alignment, spreads banks

typedef float v2f __attribute__((ext_vector_type(2)));
typedef float v8f __attribute__((ext_vector_type(8)));
typedef int   v4i __attribute__((ext_vector_type(4)));

// ---------------------------------------------------------------------------
// CDNA5 async global->LDS staging (ASYNCcnt-tracked), with fallback.
// Probe-confirmed parameter type: pointer to 'int __vector(4)' (generic AS).
// ---------------------------------------------------------------------------
#if defined(__has_builtin)
#if __has_builtin(__builtin_amdgcn_global_load_async_to_lds_b128)
#define HAVE_ASYNC_LDS 1
#endif
#endif

#ifdef HAVE_ASYNC_LDS
__device__ __forceinline__ void async_cp16(const float* g, float* l) {
    __builtin_amdgcn_global_load_async_to_lds_b128((v4i*)g, (v4i*)l, 0, 0);
}
__device__ __forceinline__ void wait_async0() {
#if __has_builtin(__builtin_amdgcn_s_wait_asynccnt)
    __builtin_amdgcn_s_wait_asynccnt(0);
#else
    asm volatile("s_wait_asynccnt 0x0" ::: "memory");
#endif
}
#else
__device__ __forceinline__ void async_cp16(const float* g, float* l) {
    *(float4*)l = *(const float4*)g;
}
__device__ __forceinline__ void wait_async0() {}
#endif

__device__ __forceinline__ v8f wmma4(v2f a, v2f b, v8f c) {
    // D = A(16x4 f32) * B(4x16 f32) + C(16x16 f32), full f32 precision
    return __builtin_amdgcn_wmma_f32_16x16x4_f32(false, a, false, b, (short)0, c,
                                                 false, false);
}

// ---------------------------------------------------------------------------
// Kernel 1: per-(b,h,c) chunk states:  states[p,n] = sum_l X[l,p]*exp(cs63-cs[l])*B[l,n]
// Also writes dA[b,h,c] = cs[63].
// ---------------------------------------------------------------------------
__global__ __launch_bounds__(128)
void ssd_chunk_states(const float* __restrict__ X, const float* __restrict__ A,
                      const float* __restrict__ Bm, float* __restrict__ states,
                      float* __restrict__ dA)
{
    __shared__ float As[LL];
    __shared__ float cs[LL];
    __shared__ float dec[LL];
    __shared__ float Xs[LL][PAD];
    __shared__ float Bs[LL][PAD];

    const int cc = blockIdx.x, hh = blockIdx.y, bb = blockIdx.z;
    const int tid  = threadIdx.x;
    const int lane = tid & 31, w = tid >> 5;
    const int mrow = lane & 15, half = lane >> 4, kb = half * 2;
    const int r0 = (w >> 1) * 32;   // this wave's 32-row strip (p)
    const int c0 = (w & 1) * 32;    // this wave's 32-col strip (n)

    // kick off async tile staging first, overlap with A-chain setup
    for (int idx = tid; idx < LL * 16; idx += 128) {
        int row = idx >> 4, q = idx & 15;
        size_t gb = (((size_t)bb * SS + cc * LL + row) * HH + hh) * (size_t)PP + q * 4;
        async_cp16(X + gb,  &Xs[row][q * 4]);
        async_cp16(Bm + gb, &Bs[row][q * 4]);
    }

    if (tid < LL) As[tid] = A[((size_t)bb * SS + cc * LL + tid) * HH + hh];
    __syncthreads();
    if (tid == 0) {
        float a = 0.f;
        for (int l = 0; l < LL; ++l) { a += As[l]; cs[l] = a; }
    }
    __syncthreads();
    if (tid < LL) dec[tid] = expf(cs[LL - 1] - cs[tid]);
    wait_async0();
    __syncthreads();

    const v8f zero8 = {0.f,0.f,0.f,0.f,0.f,0.f,0.f,0.f};
    v8f acc[2][2] = {{zero8, zero8}, {zero8, zero8}};

#pragma unroll
    for (int k0 = 0; k0 < LL; k0 += 4) {
        const int kA = k0 + kb;
        const float d0 = dec[kA], d1 = dec[kA + 1];
        v2f a0, a1, b0, b1;
        a0.x = Xs[kA][r0 + mrow]      * d0;  a0.y = Xs[kA + 1][r0 + mrow]      * d1;
        a1.x = Xs[kA][r0 + 16 + mrow] * d0;  a1.y = Xs[kA + 1][r0 + 16 + mrow] * d1;
        b0.x = Bs[kA][c0 + mrow];            b0.y = Bs[kA + 1][c0 + mrow];
        b1.x = Bs[kA][c0 + 16 + mrow];       b1.y = Bs[kA + 1][c0 + 16 + mrow];
        acc[0][0] = wmma4(a0, b0, acc[0][0]);
        acc[0][1] = wmma4(a0, b1, acc[0][1]);
        acc[1][0] = wmma4(a1, b0, acc[1][0]);
        acc[1][1] = wmma4(a1, b1, acc[1][1]);
    }

    const size_t sb = (((size_t)bb * HH + hh) * CH + cc) * (size_t)(PP * NN);
#pragma unroll
    for (int si = 0; si < 2; ++si)
#pragma unroll
        for (int sj = 0; sj < 2; ++sj)
#pragma unroll
            for (int r = 0; r < 8; ++r) {
                int p = r0 + si * 16 + r + 8 * half;
                int n = c0 + sj * 16 + mrow;
                states[sb + p * NN + n] = acc[si][sj][r];
            }

    if (tid == 0) dA[((size_t)bb * HH + hh) * CH + cc] = cs[LL - 1];
}

// ---------------------------------------------------------------------------
// Kernel 2: per-(b,h) exclusive cross-chunk scan, IN PLACE:
//   prev[c] written over states[c];  carry = (carry + states_c) * exp(dA_c)
// ---------------------------------------------------------------------------
__global__ __launch_bounds__(256)
void ssd_scan(float* __restrict__ states, const float* __restrict__ dA)
{
    const int bh = blockIdx.x;
    const size_t base = (size_t)bh * CH * (PP * NN);
    float carry[16];
#pragma unroll
    for (int j = 0; j < 16; ++j) carry[j] = 0.f;

    for (int c = 0; c < CH; ++c) {
        const float e = expf(dA[bh * CH + c]);
        const size_t cb = base + (size_t)c * (PP * NN) + threadIdx.x;
#pragma unroll
        for (int j = 0; j < 16; ++j) {
            const size_t off = cb + 256 * j;
            float t = states[off];
            states[off] = carry[j];              // prev-state entering chunk c
            carry[j] = (carry[j] + t) * e;       // matches inclusive-segsum ref
        }
    }
}

// ---------------------------------------------------------------------------
// Kernel 3: per-(b,h,c) output:
//   G = C*B^T, masked by L[i,j]=exp(cs[i]-cs[j]+A[j]) (i>=j)
//   Y = (G.L)*X + diag(exp(cs)) * (C * prev^T)
// ---------------------------------------------------------------------------
__global__ __launch_bounds__(128)
void ssd_output(const float* __restrict__ X, const float* __restrict__ A,
                const float* __restrict__ Bm, const float* __restrict__ Cm,
                const float* __restrict__ states, float* __restrict__ Y)
{
    __shared__ float As[LL];
    __shared__ float cs[LL];
    __shared__ float Csb[LL][PAD];
    __shared__ float Bsb[LL][PAD];   // reused as masked G after barrier
    __shared__ float Xs[LL][PAD];
    __shared__ float Ssb[LL][PAD];   // prev state, [p][n]

    const int cc = blockIdx.x, hh = blockIdx.y, bb = blockIdx.z;
    const int tid  = threadIdx.x;
    const int lane = tid & 31, w = tid >> 5;
    const int mrow = lane & 15, half = lane >> 4, kb = half * 2;
    const int r0 = (w >> 1) * 32;   // 32-row strip (l rows of output)
    const int c0 = (w & 1) * 32;    // 32-col strip

    const size_t sb = (((size_t)bb * HH + hh) * CH + cc) * (size_t)(PP * NN);
    for (int idx = tid; idx < LL * 16; idx += 128) {
        int row = idx >> 4, q = idx & 15;
        size_t gb = (((size_t)bb * SS + cc * LL + row) * HH + hh) * (size_t)PP + q * 4;
        async_cp16(X + gb,                &Xs[row][q * 4]);
        async_cp16(Bm + gb,               &Bsb[row][q * 4]);
        async_cp16(Cm + gb,               &Csb[row][q * 4]);
        async_cp16(states + sb + idx * 4, &Ssb[row][q * 4]);
    }

    if (tid < LL) As[tid] = A[((size_t)bb * SS + cc * LL + tid) * HH + hh];
    __syncthreads();
    if (tid == 0) {
        float a = 0.f;
        for (int l = 0; l < LL; ++l) { a += As[l]; cs[l] = a; }
    }
    wait_async0();
    __syncthreads();

    const v8f zero8 = {0.f,0.f,0.f,0.f,0.f,0.f,0.f,0.f};

    // ---- GEMM 1: G = C * B^T (contract n) ----
    v8f g[2][2] = {{zero8, zero8}, {zero8, zero8}};
#pragma unroll
    for (int k0 = 0; k0 < NN; k0 += 4) {
        const int kA = k0 + kb;
        v2f a0, a1, b0, b1;
        a0.x = Csb[r0 + mrow][kA];        a0.y = Csb[r0 + mrow][kA + 1];
        a1.x = Csb[r0 + 16 + mrow][kA];   a1.y = Csb[r0 + 16 + mrow][kA + 1];
        b0.x = Bsb[c0 + mrow][kA];        b0.y = Bsb[c0 + mrow][kA + 1];       // B^T
        b1.x = Bsb[c0 + 16 + mrow][kA];   b1.y = Bsb[c0 + 16 + mrow][kA + 1];
        g[0][0] = wmma4(a0, b0, g[0][0]);
        g[0][1] = wmma4(a0, b1, g[0][1]);
        g[1][0] = wmma4(a1, b0, g[1][0]);
        g[1][1] = wmma4(a1, b1, g[1][1]);
    }
    __syncthreads();                 // everyone done reading Bsb

    // mask with L and stash into Bsb (now "Gs")
#pragma unroll
    for (int si = 0; si < 2; ++si)
#pragma unroll
        for (int sj = 0; sj < 2; ++sj)
#pragma unroll
            for (int r = 0; r < 8; ++r) {
                int i  = r0 + si * 16 + r + 8 * half;
                int jc = c0 + sj * 16 + mrow;
                float v = (i >= jc) ? g[si][sj][r] * expf(cs[i] - cs[jc] + As[jc]) : 0.f;
                Bsb[i][jc] = v;
            }
    __syncthreads();

    // ---- GEMM 2: Y_dig = Gs * X (contract l) ----
    v8f acc[2][2] = {{zero8, zero8}, {zero8, zero8}};
#pragma unroll
    for (int k0 = 0; k0 < LL; k0 += 4) {
        const int kA = k0 + kb;
        v2f a0, a1, b0, b1;
        a0.x = Bsb[r0 + mrow][kA];        a0.y = Bsb[r0 + mrow][kA + 1];
        a1.x = Bsb[r0 + 16 + mrow][kA];   a1.y = Bsb[r0 + 16 + mrow][kA + 1];
        b0.x = Xs[kA][c0 + mrow];         b0.y = Xs[kA + 1][c0 + mrow];
        b1.x = Xs[kA][c0 + 16 + mrow];    b1.y = Xs[kA + 1][c0 + 16 + mrow];
        acc[0][0] = wmma4(a0, b0, acc[0][0]);
        acc[0][1] = wmma4(a0, b1, acc[0][1]);
        acc[1][0] = wmma4(a1, b0, acc[1][0]);
        acc[1][1] = wmma4(a1, b1, acc[1][1]);
    }

    // ---- GEMM 3: Y_off += diag(exp(cs)) * C * prev^T (contract n) ----
    const float esc0 = expf(cs[r0 + mrow]);
    const float esc1 = expf(cs[r0 + 16 + mrow]);
#pragma unroll
    for (int k0 = 0; k0 < NN; k0 += 4) {
        const int kA = k0 + kb;
        v2f a0, a1, b0, b1;
        a0.x = Csb[r0 + mrow][kA]      * esc0;  a0.y = Csb[r0 + mrow][kA + 1]      * esc0;
        a1.x = Csb[r0 + 16 + mrow][kA] * esc1;  a1.y = Csb[r0 + 16 + mrow][kA + 1] * esc1;
        b0.x = Ssb[c0 + mrow][kA];              b0.y = Ssb[c0 + mrow][kA + 1];       // S[p][n]
        b1.x = Ssb[c0 + 16 + mrow][kA];         b1.y = Ssb[c0 + 16 + mrow][kA + 1];
        acc[0][0] = wmma4(a0, b0, acc[0][0]);
        acc[0][1] = wmma4(a0, b1, acc[0][1]);
        acc[1][0] = wmma4(a1, b0, acc[1][0]);
        acc[1][1] = wmma4(a1, b1, acc[1][1]);
    }

#pragma unroll
    for (int si = 0; si < 2; ++si)
#pragma unroll
        for (int sj = 0; sj < 2; ++sj)
#pragma unroll
            for (int r = 0; r < 8; ++r) {
                int i = r0 + si * 16 + r + 8 * half;
                int p = c0 + sj * 16 + mrow;
                Y[(((size_t)bb * SS + cc * LL + i) * HH + hh) * PP + p] = acc[si][sj][r];
            }
}

// ---------------------------------------------------------------------------
extern "C" void kernel_launch(void* const* d_in, const int* in_sizes, int n_in,
                              void* d_out, int out_size, void* d_ws, size_t ws_size,
                              hipStream_t stream)
{
    (void)in_sizes; (void)n_in; (void)out_size; (void)ws_size;

    const float* X  = (const float*)d_in[0];
    const float* A  = (const float*)d_in[1];
    const float* Bm = (const float*)d_in[2];
    const float* Cm = (const float*)d_in[3];
    float* Y = (float*)d_out;

    float* states = (float*)d_ws;                               // 67 MB
    float* dA     = states + (size_t)BB * HH * CH * PP * NN;    // 16 KB

    dim3 grid(CH, HH, BB);
    ssd_chunk_states<<<grid, 128, 0, stream>>>(X, A, Bm, states, dA);
    ssd_scan<<<BB * HH, 256, 0, stream>>>(states, dA);
    ssd_output<<<grid, 128, 0, stream>>>(X, A, Bm, Cm, states, Y);
}